// KNNGraph_77232101916764
// MI455X (gfx1250) — compile-verified
//
#include <hip/hip_runtime.h>

typedef __attribute__((ext_vector_type(2))) float v2f;
typedef __attribute__((ext_vector_type(8))) float v8f;

#define N_POINTS 16384
#define DIM      128
#define KNN      16
#define NTILES   (N_POINTS / 16)

// ---------------------------------------------------------------------------
// Kernel 1: squared norms, one wave32 per row (coalesced float4 per lane).
// ---------------------------------------------------------------------------
__global__ void knn_norms_kernel(const float* __restrict__ x,
                                 float* __restrict__ sq) {
  const int gtid = blockIdx.x * blockDim.x + threadIdx.x;
  const int row  = gtid >> 5;   // wave32
  const int lane = gtid & 31;
  const float4 v = ((const float4*)(x + (size_t)row * DIM))[lane];
  float s = v.x * v.x + v.y * v.y + v.z * v.z + v.w * v.w;
#pragma unroll
  for (int o = 16; o > 0; o >>= 1) s += __shfl_xor(s, o, 32);
  if (lane == 0) sq[row] = s;
}

// ---------------------------------------------------------------------------
// Helpers for the fused WMMA + select kernel.
// ---------------------------------------------------------------------------

// Issue all 32 B-chunk loads (one 16-col tile, WMMA 4x16 f32 B layout) into a
// distinct register array so they can all be in flight simultaneously.
__device__ __forceinline__ void load_b_tile(v2f (&b)[32],
                                            const float* __restrict__ x,
                                            int jt, int r, int h) {
  const float* bp = x + (size_t)(jt * 16 + r) * DIM + 2 * h;
#pragma unroll
  for (int kk = 0; kk < 32; ++kk)
    b[kk] = *(const v2f*)(bp + 4 * kk);
}

// 32 f32 WMMAs into 4 independent accumulators (no serial C chain),
// then key epilogue + per-lane top-16 insertion.
__device__ __forceinline__ void tile_compute_select(
    const v2f (&a)[32], const v2f (&b)[32], int jt, int r, int h, int myrow,
    const float* __restrict__ sq, float (&bd)[KNN], int (&bi)[KNN],
    float (*s_key)[17]) {
  v8f acc[4];
#pragma unroll
  for (int q = 0; q < 4; ++q)
    acc[q] = (v8f){0.f, 0.f, 0.f, 0.f, 0.f, 0.f, 0.f, 0.f};

#pragma unroll
  for (int kk = 0; kk < 32; ++kk) {
    acc[kk & 3] = __builtin_amdgcn_wmma_f32_16x16x4_f32(
        /*neg_a=*/false, a[kk], /*neg_b=*/false, b[kk],
        /*c_mod=*/(short)0, acc[kk & 3], /*reuse_a=*/false, /*reuse_b=*/false);
  }
  const v8f c = (acc[0] + acc[1]) + (acc[2] + acc[3]);

  // Ranking key within a row: sq[j] - 2*dot (sq[i] row-constant, sqrt monotone).
  const float sqj = sq[jt * 16 + r];
#pragma unroll
  for (int v = 0; v < 8; ++v)
    s_key[v + 8 * h][r] = sqj - 2.0f * c[v];
  __syncthreads();

  // Lane (r,h) scans candidates h*8 .. h*8+7 of row r.
#pragma unroll
  for (int t = 0; t < 8; ++t) {
    const int   cc  = h * 8 + t;
    const float key = s_key[r][cc];
    const int   j   = jt * 16 + cc;
    if (key < bd[KNN - 1] && j != myrow) {   // exclude self (reference drops it)
      bd[KNN - 1] = key;
      bi[KNN - 1] = j;
#pragma unroll
      for (int s = KNN - 1; s > 0; --s) {
        if (bd[s] < bd[s - 1]) {
          const float td = bd[s]; bd[s] = bd[s - 1]; bd[s - 1] = td;
          const int   ti = bi[s]; bi[s] = bi[s - 1]; bi[s - 1] = ti;
        }
      }
    }
  }
  __syncthreads();
}

// ---------------------------------------------------------------------------
// Kernel 2: fused Gram-tile (f32 WMMA 16x16x4) + per-row top-16 selection.
// One wave32 per 16-row block; B tiles double-buffered in registers so the
// 32 global_load_b64 of tile jt+1 overlap the 32 WMMAs + selection of tile jt.
// ---------------------------------------------------------------------------
__launch_bounds__(32)
__global__ void knn_select_kernel(const float* __restrict__ x,
                                  const float* __restrict__ sq,
                                  int* __restrict__ out) {
  __shared__ float s_key[16][17];  // padded to reduce bank conflicts
  __shared__ float s_md[16][32];
  __shared__ int   s_mi[16][32];

  const int lane  = threadIdx.x;   // 0..31
  const int r     = lane & 15;
  const int h     = lane >> 4;     // K-pair / candidate-half select
  const int row0  = blockIdx.x * 16;
  const int myrow = row0 + r;

  // Preload A tile: 16 rows x 128 K as 32 chunks in WMMA 16x4 f32 layout.
  v2f a[32];
  {
    const float* ap = x + (size_t)myrow * DIM + 2 * h;
#pragma unroll
    for (int kk = 0; kk < 32; ++kk)
      a[kk] = *(const v2f*)(ap + 4 * kk);
  }

  float bd[KNN];
  int   bi[KNN];
#pragma unroll
  for (int t = 0; t < KNN; ++t) { bd[t] = 3.0e38f; bi[t] = -1; }

  // Software-pipelined main loop over 1024 column tiles (NTILES is even).
  v2f b0[32], b1[32];
  load_b_tile(b0, x, 0, r, h);
  for (int jt = 0; jt < NTILES; jt += 2) {
    load_b_tile(b1, x, jt + 1, r, h);                     // prefetch odd tile
    tile_compute_select(a, b0, jt, r, h, myrow, sq, bd, bi, s_key);
    if (jt + 2 < NTILES)
      load_b_tile(b0, x, jt + 2, r, h);                   // prefetch next even
    tile_compute_select(a, b1, jt + 1, r, h, myrow, sq, bd, bi, s_key);
  }

  // Merge the two per-half sorted lists of each row (2-pointer merge in LDS).
#pragma unroll
  for (int t = 0; t < KNN; ++t) {
    s_md[r][h * 16 + t] = bd[t];
    s_mi[r][h * 16 + t] = bi[t];
  }
  __syncthreads();

  if (h == 0) {
    int p0 = 0, p1 = 0;
#pragma unroll
    for (int t = 0; t < KNN; ++t) {
      const float d0 = s_md[r][p0];
      const float d1 = s_md[r][16 + p1];
      int sel;
      if (d0 <= d1) { sel = s_mi[r][p0];      ++p0; }
      else          { sel = s_mi[r][16 + p1]; ++p1; }
      out[(size_t)myrow * KNN + t] = sel;
    }
  }
}

// ---------------------------------------------------------------------------
extern "C" void kernel_launch(void* const* d_in, const int* in_sizes, int n_in,
                              void* d_out, int out_size, void* d_ws, size_t ws_size,
                              hipStream_t stream) {
  const float* x   = (const float*)d_in[0];   // [16384,128] fp32
  float*       sq  = (float*)d_ws;            // 16384 floats of scratch
  int*         out = (int*)d_out;             // [16384,16] int32 indices

  knn_norms_kernel<<<(N_POINTS * 32) / 256, 256, 0, stream>>>(x, sq);
  knn_select_kernel<<<NTILES, 32, 0, stream>>>(x, sq, out);
}